// SpikeTemperature_21629455303095
// MI455X (gfx1250) — compile-verified
//
#include <hip/hip_runtime.h>
#include <math.h>

// Problem constants (match reference: B=16, N=8192, E=512, D=128)
#define B_ 16
#define N_ 8192
#define E_ 512
#define D_ 128

typedef __attribute__((ext_vector_type(2))) float v2f;
typedef __attribute__((ext_vector_type(8))) float v8f;

// ---------------------------------------------------------------------------
// Phase 0: zero edge accumulators AND precombine weights:
//   wprep[0..D)    = W1 + W2   (combined dot weights)
//   wprep[D..2D)   = W2        (deviation weights)
// ---------------------------------------------------------------------------
__global__ void st_prep_kernel(float* __restrict__ edge_zero, int ez,
                               const float* __restrict__ W,
                               float* __restrict__ wprep) {
    int i = blockIdx.x * blockDim.x + threadIdx.x;
    if (i < ez) edge_zero[i] = 0.0f;
    if (i < D_) {
        wprep[i]      = W[i] + W[D_ + i];
        wprep[D_ + i] = W[D_ + i];
    }
}

// ---------------------------------------------------------------------------
// Phase 1: one wave32 per 16-node tile.
//   D[16x16] = A(16x4 weights) x B(4x16 obs chunk), accumulated over 32 steps.
//     A row 0 = W1+W2, row 1 = W2, rows 2..15 = 0.
//     B[k][n] = obs[node n][k0+k]  (nodes striped across lanes).
//   => D row 0 (c[0], lanes 0-15) = p_comb per node   (coalesced store)
//      D row 1 (c[1], lanes 0-15) = p2 per node       (segment-sum atomics)
// V_WMMA_F32_16X16X4_F32 layouts (ISA 7.12.2):
//   A 16x4 fp32: lane m (0-15) holds K=0,1 in vgpr0/1; lanes 16-31 hold K=2,3.
//   B 4x16 fp32: lane n (0-15) holds rows K=0,1; lanes 16-31 hold rows K=2,3.
//   C/D 16x16:   vgpr r = row r (lanes 0-15) / row r+8 (lanes 16-31), col=lane.
// All 32 weight A-fragments are prebuilt into registers before the K-loop, so
// the hot loop is exactly: 1x global_load_b64 + 1x v_wmma per K-step.
// ---------------------------------------------------------------------------
__global__ __launch_bounds__(256) void st_phase1_wmma(
        const float* __restrict__ obs,       // (B,N,D)
        const int*   __restrict__ idx,       // (B,N)
        const float* __restrict__ wprep,     // (2D): [W1+W2 | W2]
        float* __restrict__ p_comb,          // (B*N)
        float* __restrict__ edge_acc,        // (B*E)
        float* __restrict__ edge_cnt) {      // (B*E)
    const int lane  = threadIdx.x & 31;
    const int wave  = threadIdx.x >> 5;
    const int tile  = blockIdx.x * 8 + wave;     // 8 waves / block, 8192 tiles
    const int base  = tile << 4;                 // first node (flat b*N+n)
    const int l16   = lane & 15;                 // A: row M;  B: col N (node)
    const int khalf = (lane >> 4) << 1;          // K sub-offset {0,2}

    // B-matrix source: this lane's node row of obs.
    const float* rowp = obs + (size_t)(base + l16) * D_ + khalf;

    // A-matrix source: row 0 lanes read Wc, row 1 lanes read W2, rest masked.
    const bool  nz    = (l16 < 2);               // nonzero A rows
    const float* wrow = wprep + (l16 == 1 ? D_ : 0) + khalf;

    // Prebuild all 32 weight fragments (loop-invariant; ~64 VGPRs).
    v2f wfrag[D_ / 4];
    #pragma unroll
    for (int s = 0; s < D_ / 4; ++s) {
        v2f w = *(const v2f*)(wrow + 4 * s);
        wfrag[s].x = nz ? w.x : 0.0f;
        wfrag[s].y = nz ? w.y : 0.0f;
    }

    v8f c = {0.f, 0.f, 0.f, 0.f, 0.f, 0.f, 0.f, 0.f};

    // Hot loop: one obs load + one fp32 WMMA per K-step.
    #pragma unroll
    for (int s = 0; s < D_ / 4; ++s) {
        v2f b = *(const v2f*)(rowp + 4 * s);
        c = __builtin_amdgcn_wmma_f32_16x16x4_f32(
                /*neg_a=*/false, wfrag[s], /*neg_b=*/false, b,
                /*c_mod=*/(short)0, c, /*reuse_a=*/false, /*reuse_b=*/false);
    }

    // Epilogue: row 0 = p_comb (coalesced), row 1 = p2 (one atomic pair/lane).
    if (lane < 16) {
        const int node = base + lane;
        p_comb[node] = c[0];
        const int b = base / N_;                 // tile never crosses a batch
        const int e = idx[node];
        atomicAdd(&edge_acc[b * E_ + e], c[1]);
        atomicAdd(&edge_cnt[b * E_ + e], 1.0f);
    }
}

// ---------------------------------------------------------------------------
// Phase 2: per-edge scalar s_edge = acc / max(cnt, 1)   (in place over acc)
// ---------------------------------------------------------------------------
__global__ void st_phase2_edge(float* __restrict__ acc,
                               const float* __restrict__ cnt, int n) {
    int i = blockIdx.x * blockDim.x + threadIdx.x;
    if (i < n) acc[i] = acc[i] / fmaxf(cnt[i], 1.0f);
}

// ---------------------------------------------------------------------------
// Phase 3: temperature = 1 + (1 - sigmoid(p_comb - s_edge[idx] + bias)) * e^lts
// ---------------------------------------------------------------------------
__global__ void st_phase3_out(const float* __restrict__ p_comb,
                              const int*   __restrict__ idx,
                              const float* __restrict__ s_edge,
                              const float* __restrict__ bias,
                              const float* __restrict__ lts,
                              float* __restrict__ out, int n) {
    int i = blockIdx.x * blockDim.x + threadIdx.x;
    if (i >= n) return;
    const int   b  = i / N_;
    const float z  = p_comb[i] - s_edge[b * E_ + idx[i]] + bias[0];
    const float sg = 1.0f / (1.0f + expf(-z));
    const float ts = expf(lts[0]);
    out[i] = 1.0f + (1.0f - sg) * ts;
}

// ---------------------------------------------------------------------------
// Workspace (floats):
//   [0, B*N)            p_comb
//   [B*N, +B*E)         edge_acc (-> s_edge after phase 2)
//   [.., +B*E)          edge_cnt
//   [.., +2D)           wprep
// ---------------------------------------------------------------------------
extern "C" void kernel_launch(void* const* d_in, const int* in_sizes, int n_in,
                              void* d_out, int out_size, void* d_ws, size_t ws_size,
                              hipStream_t stream) {
    const float* obs  = (const float*)d_in[0];
    // d_in[1] = incidence matrix (256 MiB) — redundant with idx, never touched.
    const int*   idx  = (const int*)  d_in[2];
    const float* W    = (const float*)d_in[3];
    const float* bias = (const float*)d_in[4];
    const float* lts  = (const float*)d_in[5];
    float*       out  = (float*)d_out;

    float* p_comb   = (float*)d_ws;
    float* edge_acc = p_comb + (size_t)B_ * N_;
    float* edge_cnt = edge_acc + (size_t)B_ * E_;
    float* wprep    = edge_cnt + (size_t)B_ * E_;

    // Phase 0: zero acc+cnt (contiguous) and precombine weights.
    const int ez = 2 * B_ * E_;
    st_prep_kernel<<<(ez + 255) / 256, 256, 0, stream>>>(edge_acc, ez, W, wprep);

    // Phase 1: 8192 tiles, 8 waves per 256-thread block -> 1024 blocks.
    st_phase1_wmma<<<(B_ * N_ / 16) / 8, 256, 0, stream>>>(
        obs, idx, wprep, p_comb, edge_acc, edge_cnt);

    // Phase 2: 8192 edges.
    st_phase2_edge<<<(B_ * E_ + 255) / 256, 256, 0, stream>>>(
        edge_acc, edge_cnt, B_ * E_);

    // Phase 3: 131072 nodes.
    st_phase3_out<<<(B_ * N_ + 255) / 256, 256, 0, stream>>>(
        p_comb, idx, edge_acc, bias, lts, out, B_ * N_);
}